// EdgeConvBlock_34351148434013
// MI455X (gfx1250) — compile-verified
//
#include <hip/hip_runtime.h>

// EdgeConvBlock collapsed form:
//   out[b, i, :] = max over 256 rows of x[b].reshape(256,128) of
//                  ( relu(row @ W1 + b1) @ W2 + b2 )        (independent of i)
// One workgroup (8 wave32s) per batch b; f32 WMMA 16x16x4 for both GEMMs.
// W1/W2 staged into LDS by the Tensor Data Mover (tensor_load_to_lds).

typedef float v2f __attribute__((ext_vector_type(2)));
typedef float v8f __attribute__((ext_vector_type(8)));
typedef unsigned int u32x4 __attribute__((ext_vector_type(4)));
typedef int i32x4 __attribute__((ext_vector_type(4)));
typedef int i32x8 __attribute__((ext_vector_type(8)));

#define NB   16   // batches
#define NC   64   // channels
#define NT   512  // time
#define NOUT 128  // OUT
#define MROWS 256 // rows per batch = NC*NT/128
#define KDIM 128  // 2C
#define NDIM 128  // OUT

// Deterministic float atomic-max on LDS via int/uint ordering trick
// (valid with init = -inf for arbitrary-signed IEEE floats).
__device__ inline void atomicMaxF(float* addr, float v) {
    if (v >= 0.0f) atomicMax((int*)addr, __float_as_int(v));
    else           atomicMin((unsigned int*)addr, __float_as_uint(v));
}

// TDM 1-D copy: nelem f32 elements from global -> LDS.
// D# group0: count=1 | lds_addr | global_addr[56:0] | type=2
// D# group1: data_size=4B; tensor_dim0 = tile_dim0 = nelem; stride0 = nelem.
// Groups 2/3 (and the trailing group) zero: tensor <= 2D.
// Issued once per wave; EXEC is ignored by the TDM.
__device__ inline void tdm_copy_f32(unsigned lds_byte_off, const float* gsrc,
                                    unsigned nelem /* <= 65535 */) {
    unsigned long long ga = (unsigned long long)(uintptr_t)gsrc;
    u32x4 g0;
    g0.x = 1u;                                              // count=1 (valid, user)
    g0.y = lds_byte_off;                                    // lds_addr (bytes)
    g0.z = (unsigned)(ga & 0xFFFFFFFFu);                    // global_addr[31:0]
    g0.w = (unsigned)((ga >> 32) & 0x01FFFFFFu) | (2u << 30); // addr[56:32] | type=2
    i32x8 g1;
    g1[0] = (int)(2u << 16);                 // data_size = 4 bytes
    g1[1] = (int)((nelem & 0xFFFFu) << 16);  // tensor_dim0[15:0]  (bits 63:48)
    g1[2] = (int)((nelem >> 16) & 0xFFFFu);  // tensor_dim0[31:16] (bits 79:64)
    g1[3] = (int)((nelem & 0xFFFFu) << 16);  // tile_dim0 (bits 127:112)
    g1[4] = 0;                               // tile_dim1/2 unused
    g1[5] = (int)nelem;                      // tensor_dim0_stride[31:0]
    g1[6] = 0;
    g1[7] = 0;
    i32x4 z4 = {0, 0, 0, 0};
    i32x8 z8 = {0, 0, 0, 0, 0, 0, 0, 0};
    // 6-arg form (clang-23 / therock-10.0 toolchain)
    __builtin_amdgcn_tensor_load_to_lds(g0, g1, z4, z4, z8, 0);
}

__global__ __launch_bounds__(256)
void edgeconv_wmma_kernel(const float* __restrict__ x,
                          const float* __restrict__ W1,
                          const float* __restrict__ b1,
                          const float* __restrict__ W2,
                          const float* __restrict__ b2,
                          float* __restrict__ out) {
    __shared__ float sH[MROWS * KDIM];   // 128 KB: relu(X@W1+b1)
    __shared__ float sW1[KDIM * NDIM];   //  64 KB
    __shared__ float sW2[KDIM * NDIM];   //  64 KB
    __shared__ float sMax[NDIM];
    __shared__ float sB1[NDIM];
    __shared__ float sB2[NDIM];

    const int tid   = threadIdx.x;
    const int wid   = tid >> 5;          // wave id 0..7
    const int lane  = tid & 31;
    const int lhalf = lane >> 4;         // 0/1: selects K pair (A/B) or M half (D)
    const int lmod  = lane & 15;
    const int b     = blockIdx.x;

    // --- Stage W1/W2 into LDS with the Tensor Data Mover (wave 0 only) ---
    if (wid == 0) {
        tdm_copy_f32((unsigned)(uintptr_t)(void*)sW1, W1, KDIM * NDIM);
        tdm_copy_f32((unsigned)(uintptr_t)(void*)sW2, W2, KDIM * NDIM);
    }
    if (tid < NDIM) {
        sMax[tid] = -__builtin_inff();
        sB1[tid]  = b1[tid];
        sB2[tid]  = b2[tid];
    }
    if (wid == 0) __builtin_amdgcn_s_wait_tensorcnt(0);
    __syncthreads();

    const float* xb = x + b * (NC * NT); // [256 x 128] row-major view

    // ---------------- Layer 1: H = relu(X @ W1 + b1) ----------------
    for (int nt = 0; nt < 8; ++nt) {
        const int n = nt * 16 + lmod;            // B/D column for this lane
        v2f Bf[32];
        #pragma unroll
        for (int kk = 0; kk < 32; ++kk) {        // preload all B fragments (K=128)
            const int ka = kk * 4 + 2 * lhalf;
            Bf[kk].x = sW1[ka * NDIM + n];
            Bf[kk].y = sW1[(ka + 1) * NDIM + n];
        }
        const float bias = sB1[n];
        #pragma unroll
        for (int mi = 0; mi < 2; ++mi) {         // reuse B across both M-tiles
            const int mt   = wid * 2 + mi;
            const int arow = mt * 16 + lmod;
            const float* ap = xb + arow * KDIM + 2 * lhalf;
            v8f acc = {};
            #pragma unroll
            for (int kk = 0; kk < 32; ++kk) {
                v2f a = *(const v2f*)(ap + kk * 4);   // A[m, ka..ka+1], 8B aligned
                acc = __builtin_amdgcn_wmma_f32_16x16x4_f32(
                          false, a, false, Bf[kk], (short)0, acc, false, false);
            }
            #pragma unroll
            for (int v = 0; v < 8; ++v) {
                float h = fmaxf(acc[v] + bias, 0.0f);
                const int m = mt * 16 + v + 8 * lhalf;   // D: M = v + 8*(lane/16)
                sH[m * KDIM + n] = h;
            }
        }
    }
    __syncthreads();

    // ------------- Layer 2: O = H @ W2 + b2, then column max -------------
    for (int nt = 0; nt < 8; ++nt) {
        const int n = nt * 16 + lmod;
        v2f Bf[32];
        #pragma unroll
        for (int kk = 0; kk < 32; ++kk) {
            const int ka = kk * 4 + 2 * lhalf;
            Bf[kk].x = sW2[ka * NDIM + n];
            Bf[kk].y = sW2[(ka + 1) * NDIM + n];
        }
        const float bias = sB2[n];
        #pragma unroll
        for (int mi = 0; mi < 2; ++mi) {
            const int mt   = wid * 2 + mi;
            const int arow = mt * 16 + lmod;
            const float* hp = &sH[arow * KDIM + 2 * lhalf];
            v8f acc = {};
            #pragma unroll
            for (int kk = 0; kk < 32; ++kk) {
                v2f a = *(const v2f*)(hp + kk * 4);
                acc = __builtin_amdgcn_wmma_f32_16x16x4_f32(
                          false, a, false, Bf[kk], (short)0, acc, false, false);
            }
            float m8 = -__builtin_inff();
            #pragma unroll
            for (int v = 0; v < 8; ++v)
                m8 = fmaxf(m8, acc[v] + bias);
            // lanes l and l^16 hold the same column n (different M halves)
            m8 = fmaxf(m8, __shfl_xor(m8, 16, 32));
            if (lane < 16) atomicMaxF(&sMax[n], m8);
        }
    }
    __syncthreads();

    // ------------- Broadcast: out[b, i, :] = sMax for all i (float4) -------------
    float4* ob4 = (float4*)(out + b * (NC * NOUT));      // 2048 float4s
    const float4* sM4 = (const float4*)sMax;             // 32 float4s
    for (int idx = tid; idx < (NC * NOUT) / 4; idx += 256)
        ob4[idx] = sM4[idx & 31];
}

extern "C" void kernel_launch(void* const* d_in, const int* in_sizes, int n_in,
                              void* d_out, int out_size, void* d_ws, size_t ws_size,
                              hipStream_t stream) {
    const float* x  = (const float*)d_in[0];  // [16, 64, 512]
    const float* W1 = (const float*)d_in[1];  // [128, 128]
    const float* b1 = (const float*)d_in[2];  // [128]
    const float* W2 = (const float*)d_in[3];  // [128, 128]
    const float* b2 = (const float*)d_in[4];  // [128]
    float* out = (float*)d_out;               // [16, 64, 128]

    edgeconv_wmma_kernel<<<NB, 256, 0, stream>>>(x, W1, b1, W2, b2, out);
}